// LinHRULayer_19318762898000
// MI455X (gfx1250) — compile-verified
//
#include <hip/hip_runtime.h>
#include <stdint.h>

#define L_SEQ 8192
#define H_DIM 1024
#define P_DIM 1024
#define CHUNK 128
#define NCH   (L_SEQ / CHUNK)   // 64 chunks
#define LOG2_CHUNK 7            // 2^7 = 128

// GEMM tiling: block = 8 waves -> 64x128 block tile, wave = 32x32 (2x2 WMMA)
#define KT        32            // K tile in floats
#define M_BLK     64
#define N_BLK     128
#define ASTRIDE   34            // padded LDS row stride (floats): 8B aligned, bank-spread
#define BSTRIDE   34
#define A_TILE_F  (M_BLK * ASTRIDE)   // 2176 floats
#define B_TILE_F  (N_BLK * BSTRIDE)   // 4352 floats

typedef __attribute__((ext_vector_type(2))) float v2f;
typedef __attribute__((ext_vector_type(8))) float v8f;
typedef int b64vec __attribute__((vector_size(8)));   // matches builtin's V2i

// ---------------------------------------------------------------------------
// CDNA5 async global->LDS copy (ASYNCcnt-tracked), with safe fallback.
// Builtin signature (from hipcc diagnostic): (V2i as(1)*, V2i as(3)*, Ii, Ii).
// Flat shared-address low 32 bits are the LDS byte offset (ISA 10.2 aperture
// rules), so the inttoptr truncation yields a valid as(3) pointer.
// ---------------------------------------------------------------------------
#if defined(__has_builtin)
#if __has_builtin(__builtin_amdgcn_global_load_async_to_lds_b64) && \
    __has_builtin(__builtin_amdgcn_s_wait_asynccnt)
#define USE_ASYNC_LDS 1
#endif
#endif
#ifndef USE_ASYNC_LDS
#define USE_ASYNC_LDS 0
#endif

#if USE_ASYNC_LDS
#define COPY_B64(g, l)                                                          \
  __builtin_amdgcn_global_load_async_to_lds_b64(                                \
      (__attribute__((address_space(1))) b64vec*)(uintptr_t)(g),                \
      (__attribute__((address_space(3))) b64vec*)(uint32_t)(uintptr_t)(l), 0, 0)
#define WAIT_ASYNC() __builtin_amdgcn_s_wait_asynccnt(0)
#else
#define COPY_B64(g, l) (*(v2f*)(l) = *(const v2f*)(g))
#define WAIT_ASYNC() ((void)0)
#endif

// ---------------------------------------------------------------------------
// WMMA helper: D = A(16x4 f32) * B(4x16 f32) + C, exact fp32 matrix op.
// ---------------------------------------------------------------------------
__device__ __forceinline__ v8f wmma_f32(v2f a, v2f b, v8f c) {
  return __builtin_amdgcn_wmma_f32_16x16x4_f32(false, a, false, b, (short)0, c,
                                               false, false);
}

// ---------------------------------------------------------------------------
// GEMM (NT): Out[m,n] = sum_k A[m,k] * Bm[n,k]  (row-major, K contiguous).
// Block tile 64x128 staged in LDS (double buffered); wave tile 32x32.
// Fragment layout per ISA 7.12.2 (f32 16x16x4):
//   A: lane = m + 16*g -> {A[m,k+2g], A[m,k+2g+1]}
//   B: lane = n + 16*g -> {Bm[n,k+2g], Bm[n,k+2g+1]}
//   C/D: VGPR v: lanes 0-15 (M=v, N=lane), lanes 16-31 (M=v+8, N=lane-16)
// ---------------------------------------------------------------------------
template <bool FUSE_UD>
__global__ __launch_bounds__(256)
void gemm_nt_wmma(const float* __restrict__ A, const float* __restrict__ Bm,
                  float* __restrict__ Out, int Ndim, int K,
                  const float* __restrict__ U, const float* __restrict__ Dv) {
  __shared__ float smem[2][A_TILE_F + B_TILE_F];   // 52224 bytes

  const int tid  = threadIdx.x;
  const int lane = tid & 31;
  const int wave = tid >> 5;
  const int nbn  = Ndim / N_BLK;
  const int bm0  = (blockIdx.x / nbn) * M_BLK;
  const int bn0  = (blockIdx.x % nbn) * N_BLK;

  const int wm0 = (wave >> 2) * 32;   // wave row offset inside block tile
  const int wn0 = (wave & 3) * 32;    // wave col offset inside block tile
  const int l15 = lane & 15;
  const int kh  = lane >> 4;          // K-half select (lanes 16-31 -> K+2)

  // Stage one 64xKT A panel + 128xKT B panel into smem[buf] at column kk.
  auto stage = [&](int buf, int kk) {
    float* base = &smem[buf][0];
#pragma unroll
    for (int j = 0; j < 4; ++j) {           // A: 64x32 = 1024 float2
      const int idx = tid + j * 256;
      const int row = idx >> 4;
      const int col = (idx & 15) * 2;
      const float* g = A + (size_t)(bm0 + row) * K + kk + col;
      float* l = base + row * ASTRIDE + col;
      COPY_B64(g, l);
    }
#pragma unroll
    for (int j = 0; j < 8; ++j) {           // B: 128x32 = 2048 float2
      const int idx = tid + j * 256;
      const int row = idx >> 4;
      const int col = (idx & 15) * 2;
      const float* g = Bm + (size_t)(bn0 + row) * K + kk + col;
      float* l = base + A_TILE_F + row * BSTRIDE + col;
      COPY_B64(g, l);
    }
  };

  v8f acc00 = {}, acc01 = {}, acc10 = {}, acc11 = {};

  stage(0, 0);
  const int nk = K / KT;                    // 32 iterations
  for (int t = 0; t < nk; ++t) {
    const int buf = t & 1;
    WAIT_ASYNC();          // this wave's copies into smem[buf] are done
    __syncthreads();       // ...and every other wave's too; prev compute done

    if (t + 1 < nk) stage(buf ^ 1, (t + 1) * KT);

    const float* As0 = &smem[buf][0] + (wm0 + l15) * ASTRIDE + 2 * kh;
    const float* As1 = As0 + 16 * ASTRIDE;
    const float* Bs0 = &smem[buf][0] + A_TILE_F + (wn0 + l15) * BSTRIDE + 2 * kh;
    const float* Bs1 = Bs0 + 16 * BSTRIDE;
#pragma unroll
    for (int k = 0; k < KT; k += 4) {
      v2f av0 = *(const v2f*)(As0 + k);
      v2f av1 = *(const v2f*)(As1 + k);
      v2f bv0 = *(const v2f*)(Bs0 + k);
      v2f bv1 = *(const v2f*)(Bs1 + k);
      acc00 = wmma_f32(av0, bv0, acc00);
      acc01 = wmma_f32(av0, bv1, acc01);
      acc10 = wmma_f32(av1, bv0, acc10);
      acc11 = wmma_f32(av1, bv1, acc11);
    }
  }

  auto store16 = [&](v8f acc, int mbase, int nbase) {
    const int n = bn0 + nbase + l15;
    float d = 0.0f;
    if constexpr (FUSE_UD) d = Dv[n];
#pragma unroll
    for (int v = 0; v < 8; ++v) {
      const int m = bm0 + mbase + v + 8 * kh;
      const size_t idx = (size_t)m * Ndim + n;
      float val = acc[v];
      if constexpr (FUSE_UD) val = fmaf(U[idx], d, val);
      Out[idx] = val;
    }
  };
  store16(acc00, wm0, wn0);
  store16(acc01, wm0, wn0 + 16);
  store16(acc10, wm0 + 16, wn0);
  store16(acc11, wm0 + 16, wn0 + 16);
}

// ---------------------------------------------------------------------------
// IMEX coefficients per channel (time-invariant 2x2 transition).
// ---------------------------------------------------------------------------
struct Coef { float Ac, Bc, Cc, Dc, dt, h2; };

__device__ __forceinline__ Coef coeffs(const float* __restrict__ A_diag,
                                       const float* __restrict__ steps, int p) {
  const float A  = fmaxf(A_diag[p], 0.0f);
  const float dt = 1.0f / (1.0f + __expf(-steps[p]));
  const float dt2 = dt * dt;
  Coef c;
  c.Ac = 1.0f - 0.5f * dt2 * A;
  c.Bc = -dt * A;
  c.Cc = dt * (1.0f - 0.25f * dt2 * A);
  c.Dc = c.Ac;
  c.dt = dt;
  c.h2 = 0.5f * dt2;
  return c;
}

// Pass A: per-chunk local scan from zero state; record chunk-final states.
__global__ __launch_bounds__(256)
void scan_local(const float* __restrict__ Bu, const float* __restrict__ A_diag,
                const float* __restrict__ steps,
                float* __restrict__ loc1, float* __restrict__ loc2) {
  const int tid = blockIdx.x * blockDim.x + threadIdx.x;  // NCH*P threads
  const int p = tid & (P_DIM - 1);
  const int c = tid >> 10;
  const Coef cf = coeffs(A_diag, steps, p);
  float s1 = 0.0f, s2 = 0.0f;
  const float* bp = Bu + (size_t)c * CHUNK * P_DIM + p;
  for (int t = 0; t < CHUNK; ++t) {
    const float bu = bp[(size_t)t * P_DIM];
    const float ns1 = fmaf(cf.Ac, s1, fmaf(cf.Bc, s2, bu * cf.dt));
    const float ns2 = fmaf(cf.Cc, s1, fmaf(cf.Dc, s2, bu * cf.h2));
    s1 = ns1; s2 = ns2;
  }
  loc1[c * P_DIM + p] = s1;
  loc2[c * P_DIM + p] = s2;
}

// Pass B: sequential carry combine across chunks; M^CHUNK via repeated squaring.
__global__ __launch_bounds__(256)
void scan_carry(const float* __restrict__ A_diag, const float* __restrict__ steps,
                const float* __restrict__ loc1, const float* __restrict__ loc2,
                float* __restrict__ pre1, float* __restrict__ pre2) {
  const int p = blockIdx.x * blockDim.x + threadIdx.x;    // P_DIM threads
  const Coef cf = coeffs(A_diag, steps, p);
  float a = cf.Ac, b = cf.Bc, c = cf.Cc, d = cf.Dc;
#pragma unroll
  for (int i = 0; i < LOG2_CHUNK; ++i) {                  // M -> M^128
    const float na = fmaf(a, a, b * c);
    const float nb = fmaf(a, b, b * d);
    const float nc = fmaf(c, a, d * c);
    const float nd = fmaf(c, b, d * d);
    a = na; b = nb; c = nc; d = nd;
  }
  float s1 = 0.0f, s2 = 0.0f;
  for (int ch = 0; ch < NCH; ++ch) {
    pre1[ch * P_DIM + p] = s1;                            // incoming state
    pre2[ch * P_DIM + p] = s2;
    const float l1 = loc1[ch * P_DIM + p];
    const float l2 = loc2[ch * P_DIM + p];
    const float ns1 = fmaf(a, s1, fmaf(b, s2, l1));
    const float ns2 = fmaf(c, s1, fmaf(d, s2, l2));
    s1 = ns1; s2 = ns2;
  }
}

// Pass C: re-scan each chunk seeded with its carry, write ys (position comp).
__global__ __launch_bounds__(256)
void scan_final(const float* __restrict__ Bu, const float* __restrict__ A_diag,
                const float* __restrict__ steps,
                const float* __restrict__ pre1, const float* __restrict__ pre2,
                float* __restrict__ ys) {
  const int tid = blockIdx.x * blockDim.x + threadIdx.x;
  const int p = tid & (P_DIM - 1);
  const int c = tid >> 10;
  const Coef cf = coeffs(A_diag, steps, p);
  float s1 = pre1[c * P_DIM + p];
  float s2 = pre2[c * P_DIM + p];
  const float* bp = Bu + (size_t)c * CHUNK * P_DIM + p;
  float* yp = ys + (size_t)c * CHUNK * P_DIM + p;
  for (int t = 0; t < CHUNK; ++t) {
    const float bu = bp[(size_t)t * P_DIM];
    const float ns1 = fmaf(cf.Ac, s1, fmaf(cf.Bc, s2, bu * cf.dt));
    const float ns2 = fmaf(cf.Cc, s1, fmaf(cf.Dc, s2, bu * cf.h2));
    s1 = ns1; s2 = ns2;
    yp[(size_t)t * P_DIM] = s2;
  }
}

// ---------------------------------------------------------------------------
// Host launcher
// ---------------------------------------------------------------------------
extern "C" void kernel_launch(void* const* d_in, const int* in_sizes, int n_in,
                              void* d_out, int out_size, void* d_ws, size_t ws_size,
                              hipStream_t stream) {
  const float* U      = (const float*)d_in[0];  // (L, H)
  const float* A_diag = (const float*)d_in[1];  // (P,)
  const float* Bmat   = (const float*)d_in[2];  // (P, H)
  const float* Cmat   = (const float*)d_in[3];  // (H, P)
  const float* Dvec   = (const float*)d_in[4];  // (H,)
  const float* steps  = (const float*)d_in[5];  // (P,)
  float* out = (float*)d_out;                   // (L, H)

  float* ws   = (float*)d_ws;
  float* Bu   = ws;                             // L*P floats (32 MB)
  float* ys   = Bu + (size_t)L_SEQ * P_DIM;     // L*P floats (32 MB)
  float* loc1 = ys + (size_t)L_SEQ * P_DIM;     // NCH*P
  float* loc2 = loc1 + NCH * P_DIM;
  float* pre1 = loc2 + NCH * P_DIM;
  float* pre2 = pre1 + NCH * P_DIM;

  // GEMM1: Bu = U @ B^T   (M=L, N=P, K=H)
  {
    const int blocks = (L_SEQ / M_BLK) * (P_DIM / N_BLK);   // 1024
    gemm_nt_wmma<false><<<blocks, 256, 0, stream>>>(
        U, Bmat, Bu, P_DIM, H_DIM, nullptr, nullptr);
  }

  // Chunked parallel scan over the sequence dimension.
  scan_local<<<(NCH * P_DIM) / 256, 256, 0, stream>>>(Bu, A_diag, steps, loc1, loc2);
  scan_carry<<<P_DIM / 256, 256, 0, stream>>>(A_diag, steps, loc1, loc2, pre1, pre2);
  scan_final<<<(NCH * P_DIM) / 256, 256, 0, stream>>>(Bu, A_diag, steps, pre1, pre2, ys);

  // GEMM2: out = ys @ C^T + U * D   (M=L, N=H, K=P), fused epilogue
  {
    const int blocks = (L_SEQ / M_BLK) * (H_DIM / N_BLK);   // 1024
    gemm_nt_wmma<true><<<blocks, 256, 0, stream>>>(
        ys, Cmat, out, H_DIM, P_DIM, U, Dvec);
  }
}